// GPNCNN_16982300688596
// MI455X (gfx1250) — compile-verified
//
#include <hip/hip_runtime.h>
#include <hip/hip_bf16.h>

// Problem constants (from reference)
#define B_    4
#define M_    512
#define N_    512
#define CIN_  32
#define KS_   7
#define KC_   224   // KS*CIN
#define OUTP_ 64
#define OUTD_ 48
#define NQ_   50

// Split-K over the N reduction to fill the MI455X's WGPs:
// grid for k_mid = (M/16) * B * NSPLIT = 512 workgroups.
#define NSPLIT_ 4
#define NT_PER_SPLIT_ ((N_/16)/NSPLIT_)           // 8 n-tiles per split
#define PART_ ((size_t)B_*M_*KC_)                 // one partial mid buffer

typedef __attribute__((ext_vector_type(2))) float v2f;
typedef __attribute__((ext_vector_type(8))) float v8f;

__device__ __forceinline__ float softplusf(float x) {
  return fmaxf(x, 0.0f) + log1pf(__expf(-fabsf(x)));
}

__device__ __forceinline__ v8f wmma_f32(v2f a, v2f b, v8f c) {
  // D(16x16 f32) = A(16x4 f32) * B(4x16 f32) + C   -- full fp32 path
  return __builtin_amdgcn_wmma_f32_16x16x4_f32(
      /*neg_a=*/false, a, /*neg_b=*/false, b,
      /*c_mod=*/(short)0, c, /*reuse_a=*/false, /*reuse_b=*/false);
}

// Acklam's rational approximation to the inverse normal CDF
__device__ float ndtri_f(float p) {
  const float a1=-3.969683028665376e+01f, a2= 2.209460984245205e+02f,
              a3=-2.759285104469687e+02f, a4= 1.383577518672690e+02f,
              a5=-3.066479806614716e+01f, a6= 2.506628277459239e+00f;
  const float b1=-5.447609879822406e+01f, b2= 1.615858368580409e+02f,
              b3=-1.556989798598866e+02f, b4= 6.680131188771972e+01f,
              b5=-1.328068155288572e+01f;
  const float c1=-7.784894002430293e-03f, c2=-3.223964580411365e-01f,
              c3=-2.400758277161838e+00f, c4=-2.549732539343734e+00f,
              c5= 4.374664141464968e+00f, c6= 2.938163982698783e+00f;
  const float d1= 7.784695709041462e-03f, d2= 3.224671290700398e-01f,
              d3= 2.445134137142996e+00f, d4= 3.754408661907416e+00f;
  const float plow = 0.02425f;
  float x;
  if (p < plow) {
    float q = sqrtf(-2.0f * logf(p));
    x = (((((c1*q+c2)*q+c3)*q+c4)*q+c5)*q+c6) /
        ((((d1*q+d2)*q+d3)*q+d4)*q + 1.0f);
  } else if (p <= 1.0f - plow) {
    float q = p - 0.5f, r = q*q;
    x = (((((a1*r+a2)*r+a3)*r+a4)*r+a5)*r+a6)*q /
        (((((b1*r+b2)*r+b3)*r+b4)*r+b5)*r + 1.0f);
  } else {
    float q = sqrtf(-2.0f * logf(1.0f - p));
    x = -(((((c1*q+c2)*q+c3)*q+c4)*q+c5)*q+c6) /
         ((((d1*q+d2)*q+d3)*q+d4)*q + 1.0f);
  }
  return x;
}

__global__ void k_qtab(float* __restrict__ qt) {
  int i = threadIdx.x;
  if (i < NQ_) qt[i] = ndtri_f((i + 0.5f) / (float)NQ_);
}

// ---------------------------------------------------------------------------
// Kernel 1: fused kernel-matrix generation + batched GEMM via f32 WMMA.
// One workgroup = one (batch, 16-row m-tile, n-split). 7 waves = 7 channels.
// ---------------------------------------------------------------------------
__global__ __launch_bounds__(224) void k_mid(
    const float* __restrict__ xm, const float* __restrict__ xn,
    const float* __restrict__ fm, const float* __restrict__ fs,
    const float* __restrict__ lsr, const float* __restrict__ blr,
    float* __restrict__ mid_mean, float* __restrict__ mid_var)
{
  __shared__ float sKm[KS_][16][16];   // KiK tiles        (sigma1)
  __shared__ float sKv[KS_][16][16];   // Kdiag^2 tiles    (sigma2)
  __shared__ float sF[16][CIN_];       // f_mean tile
  __shared__ float sS[16][CIN_];       // f_std^2 tile

  const int b    = blockIdx.y;
  const int m0   = blockIdx.x * 16;
  const int zs   = blockIdx.z;      // n-split index
  const int tid  = threadIdx.x;
  const int lane = tid & 31;
  const int k    = tid >> 5;        // wave id == kernel channel (0..6)
  const int half = lane >> 4;
  const int ml   = lane & 15;

  const float ls   = softplusf(lsr[0]);
  const float bl   = softplusf(blr[0]);
  const float s1sq = ls*ls + bl*bl;
  const float s2sq = ls*ls + 2.0f*bl*bl;
  const float i1   = 1.0f / s1sq;
  const float i2   = 1.0f / s2sq;

  v8f aM0 = {}, aM1 = {}, aV0 = {}, aV1 = {};

  const int nt0 = zs * NT_PER_SPLIT_;
  for (int nt = nt0; nt < nt0 + NT_PER_SPLIT_; ++nt) {
    const int n0 = nt * 16;
    __syncthreads();   // previous phase-2 LDS reads done

    // Phase 1: build 16x16 kernel tiles for all 7 channels + F tiles
    for (int idx = tid; idx < 256; idx += 224) {
      const int mi = idx >> 4, ni = idx & 15;
      const float dx0 = xm[(b*M_ + m0 + mi)*2 + 0] - xn[(b*N_ + n0 + ni)*2 + 0];
      const float dx1 = xm[(b*M_ + m0 + mi)*2 + 1] - xn[(b*N_ + n0 + ni)*2 + 1];
      const float r2  = dx0*dx0 + dx1*dx1;
      const float k1  = __expf(-0.5f * r2 * i1);
      const float k2  = __expf(-0.5f * r2 * i2);
      // KiK channels (sigma1): [k, k*grad(2), k*hess(4)]
      const float t0 = dx0*dx0*i1, t1 = dx1*dx1*i1;
      sKm[0][mi][ni] = k1;
      sKm[1][mi][ni] = -k1*dx0*i1;
      sKm[2][mi][ni] = -k1*dx1*i1;
      sKm[3][mi][ni] =  k1*(t0 - 1.0f)*i1;
      sKm[4][mi][ni] =  k1*dx0*dx1*i1*i1;
      sKm[5][mi][ni] =  k1*dx0*dx1*i1*i1;
      sKm[6][mi][ni] =  k1*(t1 - 1.0f)*i1;
      // Kdiag channels (sigma2), squared (reference uses Kdiag**2)
      const float u0 = dx0*dx0*i2, u1 = dx1*dx1*i2;   // x_i^2
      const float kd0  = k2;
      const float kd1  = k2*(1.0f - u0)*i2;           // -d2[0,0]*k
      const float kd2  = k2*(1.0f - u1)*i2;           // -d2[1,1]*k
      const float kd3  = k2*(u0*u0 - 24.0f*u0 + 3.0f)*i2*i2;           // d4[0,0,0,0]
      const float kd45 = k2*(u0*u1 + 1.0f - 4.0f*(u0 + u1))*i2*i2;     // d4[0,1,0,1]
      const float kd6  = k2*(u1*u1 - 24.0f*u1 + 3.0f)*i2*i2;           // d4[1,1,1,1]
      sKv[0][mi][ni] = kd0*kd0;
      sKv[1][mi][ni] = kd1*kd1;
      sKv[2][mi][ni] = kd2*kd2;
      sKv[3][mi][ni] = kd3*kd3;
      sKv[4][mi][ni] = kd45*kd45;
      sKv[5][mi][ni] = kd45*kd45;
      sKv[6][mi][ni] = kd6*kd6;
    }
    for (int idx = tid; idx < 16*CIN_; idx += 224) {
      const int ni = idx >> 5, c = idx & 31;
      sF[ni][c] = fm[(b*N_ + n0 + ni)*CIN_ + c];
      const float s = fs[(b*N_ + n0 + ni)*CIN_ + c];
      sS[ni][c] = s * s;
    }
    __syncthreads();

    // Phase 2: each wave accumulates its channel via 16x16x4 f32 WMMA.
    // A 16x4 f32 layout: lanes0-15 M=0..15, VGPR0=K(2*half), VGPR1=K(2*half+1)
#pragma unroll
    for (int sub = 0; sub < 4; ++sub) {
      const int kk = sub*4 + half*2;     // n-index inside 16-tile
      v2f am, av, b0, b1, s0, s1;
      am.x = sKm[k][ml][kk];  am.y = sKm[k][ml][kk+1];
      av.x = sKv[k][ml][kk];  av.y = sKv[k][ml][kk+1];
      b0.x = sF[kk+0][ml];       b0.y = sF[kk+1][ml];
      b1.x = sF[kk+0][16 + ml];  b1.y = sF[kk+1][16 + ml];
      s0.x = sS[kk+0][ml];       s0.y = sS[kk+1][ml];
      s1.x = sS[kk+0][16 + ml];  s1.y = sS[kk+1][16 + ml];
      aM0 = wmma_f32(am, b0, aM0);
      aM1 = wmma_f32(am, b1, aM1);
      aV0 = wmma_f32(av, s0, aV0);
      aV1 = wmma_f32(av, s1, aV1);
    }
  }

  // C/D layout: VGPR r -> rows r (lanes0-15) and r+8 (lanes16-31), col = lane%16
  float* mm_part = mid_mean + (size_t)zs * PART_;
  float* mv_part = mid_var  + (size_t)zs * PART_;
#pragma unroll
  for (int r = 0; r < 8; ++r) {
    const int row  = m0 + r + 8*half;
    const int base = (b*M_ + row)*KC_ + k*CIN_;
    mm_part[base + ml]      = aM0[r];
    mm_part[base + 16 + ml] = aM1[r];
    mv_part[base + ml]      = aV0[r];
    mv_part[base + 16 + ml] = aV1[r];
  }
}

// ---------------------------------------------------------------------------
// Kernel 2: mid (224) -> out (64) projection, mean via W, var via W^2, WMMA.
// One wave per 16-row m-tile; sums the NSPLIT partial mid buffers on load.
// ---------------------------------------------------------------------------
__global__ __launch_bounds__(32) void k_out(
    const float* __restrict__ mid_mean, const float* __restrict__ mid_var,
    const float* __restrict__ W, const float* __restrict__ bvec,
    float* __restrict__ om, float* __restrict__ os)
{
  const int g    = blockIdx.x;
  const int b    = g / (M_/16);
  const int m0   = (g % (M_/16)) * 16;
  const int lane = threadIdx.x;
  const int half = lane >> 4;
  const int ml   = lane & 15;

  v8f aM[4] = {};
  v8f aV[4] = {};

  for (int kc = 0; kc < KC_; kc += 4) {
    const size_t arow = (size_t)(b*M_ + m0 + ml)*KC_ + kc + 2*half;
    float m0v = 0.0f, m1v = 0.0f, v0v = 0.0f, v1v = 0.0f;
#pragma unroll
    for (int p = 0; p < NSPLIT_; ++p) {
      m0v += mid_mean[p*PART_ + arow];
      m1v += mid_mean[p*PART_ + arow + 1];
      v0v += mid_var [p*PART_ + arow];
      v1v += mid_var [p*PART_ + arow + 1];
    }
    v2f am; am.x = m0v; am.y = m1v;
    v2f av; av.x = v0v; av.y = v1v;
#pragma unroll
    for (int cb = 0; cb < 4; ++cb) {
      const int j = cb*16 + ml;
      const float w0 = W[j*KC_ + kc + 2*half];
      const float w1 = W[j*KC_ + kc + 2*half + 1];
      v2f bw;  bw.x  = w0;     bw.y  = w1;
      v2f bw2; bw2.x = w0*w0;  bw2.y = w1*w1;
      aM[cb] = wmma_f32(am, bw,  aM[cb]);
      aV[cb] = wmma_f32(av, bw2, aV[cb]);
    }
  }

#pragma unroll
  for (int cb = 0; cb < 4; ++cb) {
    const int j  = cb*16 + ml;
    const float bj = bvec[j];
#pragma unroll
    for (int r = 0; r < 8; ++r) {
      const int row = m0 + r + 8*half;
      om[(b*M_ + row)*OUTP_ + j] = aM[cb][r] + bj;
      os[(b*M_ + row)*OUTP_ + j] = sqrtf(fmaxf(aV[cb][r], 1e-12f));
    }
  }
}

// ---------------------------------------------------------------------------
// Kernel 3: gating. 32 threads per (b,m): 16 scalar channels + 16 vec groups.
// ---------------------------------------------------------------------------
__global__ __launch_bounds__(256) void k_gate(
    const float* __restrict__ om, const float* __restrict__ os,
    const float* __restrict__ qt, float* __restrict__ out)
{
  const int gid = blockIdx.x * blockDim.x + threadIdx.x;
  const int bm  = gid >> 5;
  const int u   = gid & 31;
  if (bm >= B_*M_) return;
  const float INV_SQRT2PI = 0.3989422804014327f;

  if (u < 16) {
    // scalar channel: gated-ReLU moments
    const float mean = om[bm*OUTP_ + u];
    const float sd   = os[bm*OUTP_ + u];
    const float z    = mean / sd;
    const float phi  = 0.5f * (1.0f + erff(z * 0.7071067811865475f));
    const float dens = __expf(-0.5f*z*z) * INV_SQRT2PI;
    const float rm   = mean*phi + sd*dens;
    const float rv   = (mean*mean + sd*sd)*phi + mean*sd*dens - rm*rm;
    out[bm*96 + u]      = rm;
    out[bm*96 + 48 + u] = sqrtf(fmaxf(rv, 1e-6f));
  } else {
    // vector group v: sigmoid gate via 50-pt quantile quadrature
    const int v = u - 16;
    const float gmean = om[bm*OUTP_ + OUTD_ + v];
    const float gstd  = os[bm*OUTP_ + OUTD_ + v];
    float s1 = 0.0f, s2 = 0.0f;
    for (int i = 0; i < NQ_; ++i) {
      const float sg = 1.0f / (1.0f + __expf(-(gmean + gstd*qt[i])));
      s1 += sg; s2 += sg*sg;
    }
    const float smean = s1 * (1.0f/NQ_);
    const float svar  = s2 * (1.0f/NQ_) - smean*smean;
    const float mv0 = om[bm*OUTP_ + 16 + 2*v];
    const float mv1 = om[bm*OUTP_ + 16 + 2*v + 1];
    const float sv0 = os[bm*OUTP_ + 16 + 2*v];
    const float sv1 = os[bm*OUTP_ + 16 + 2*v + 1];
    const float agg = 0.5f*(sv0*sv0 + sv1*sv1);
    const float mm  = 0.5f*(mv0*mv0 + mv1*mv1);
    const float ov  = svar*agg + svar*mm + smean*smean*agg;
    const float gsd = sqrtf(fmaxf(ov, 1e-6f));
    out[bm*96 + 16 + 2*v]          = mv0*smean;
    out[bm*96 + 16 + 2*v + 1]      = mv1*smean;
    out[bm*96 + 48 + 16 + 2*v]     = gsd;
    out[bm*96 + 48 + 16 + 2*v + 1] = gsd;
  }
}

extern "C" void kernel_launch(void* const* d_in, const int* in_sizes, int n_in,
                              void* d_out, int out_size, void* d_ws, size_t ws_size,
                              hipStream_t stream) {
  const float* xm  = (const float*)d_in[0];
  const float* xn  = (const float*)d_in[1];
  const float* fm  = (const float*)d_in[2];
  const float* fs  = (const float*)d_in[3];
  const float* W   = (const float*)d_in[4];
  const float* bv  = (const float*)d_in[5];
  const float* lsr = (const float*)d_in[6];
  const float* blr = (const float*)d_in[7];

  float* ws       = (float*)d_ws;
  float* qt       = ws;                                   // 64 (padded)
  float* mid_mean = ws + 64;                              // NSPLIT * B*M*KC
  float* mid_var  = mid_mean + NSPLIT_*PART_;             // NSPLIT * B*M*KC
  float* om       = mid_var  + NSPLIT_*PART_;             // B*M*64
  float* os       = om       + (size_t)B_*M_*OUTP_;       // B*M*64
  float* out      = (float*)d_out;

  k_qtab<<<1, 64, 0, stream>>>(qt);

  dim3 g1(M_/16, B_, NSPLIT_);
  k_mid<<<g1, 224, 0, stream>>>(xm, xn, fm, fs, lsr, blr, mid_mean, mid_var);

  k_out<<<B_*(M_/16), 32, 0, stream>>>(mid_mean, mid_var, W, bv, om, os);

  k_gate<<<(B_*M_*32 + 255)/256, 256, 0, stream>>>(om, os, qt, out);
}